// ChamferLoss_76158360092798
// MI455X (gfx1250) — compile-verified
//
#include <hip/hip_runtime.h>
#include <float.h>

typedef __attribute__((ext_vector_type(2))) float v2f;
typedef __attribute__((ext_vector_type(8))) float v8f;

#define NPTS 8192
#define BATCH 4
#define MCHUNK 2048          // p2 points staged in LDS per pass (32 KB as float4)
#define NROWS_PER_WG 128     // 8 waves * 16 rows
#define NTILES (MCHUNK / 16)
#define PF 4                 // prefetch distance == unroll factor

// ---------------- init: seed global column-min array with +FLT_MAX bits ----
__global__ __launch_bounds__(256) void chamfer_init(unsigned* __restrict__ g_colmin) {
    int i = blockIdx.x * 256 + threadIdx.x;
    if (i < BATCH * NPTS) g_colmin[i] = 0x7F7FFFFFu;   // bits of +FLT_MAX
}

// ---------------- main: WMMA-tiled pairwise distances + min reductions -----
// dist[n,m] = ||p1_n||^2 + ||p2_m||^2 - 2 p1_n.p2_m computed as ONE wmma:
//   A row = (-2x1, -2y1, -2z1, 1), B col = (x2, y2, z2, ||p2||^2), C = ||p1||^2
__global__ __launch_bounds__(256) void chamfer_main(const float* __restrict__ p1,
                                                    const float* __restrict__ p2,
                                                    float*       __restrict__ g_rowmin,
                                                    unsigned*    __restrict__ g_colmin) {
    __shared__ float4   s_p2[MCHUNK];      // (x, y, z, ||p2||^2)
    __shared__ unsigned s_colmin[NPTS];    // workgroup-local col mins (uint-ordered)

    const int tid  = threadIdx.x;
    const int lane = tid & 31;
    const int wave = tid >> 5;
    const int half = lane >> 4;            // 0 -> K={0,1}, 1 -> K={2,3}
    const int col  = lane & 15;

    const int b      = blockIdx.x >> 6;    // 4 batches
    const int nchunk = blockIdx.x & 63;    // 64 row-chunks of 128
    const int nbase  = nchunk * NROWS_PER_WG + wave * 16;

    // ---- A tile: this lane's p1 row (M = nbase+col), K-pair chosen by half ----
    const float* p1r = p1 + ((size_t)b * NPTS + (size_t)(nbase + col)) * 3;
    float x1 = p1r[0], y1 = p1r[1], z1 = p1r[2];
    float n1_lane = x1 * x1 + y1 * y1 + z1 * z1;

    v2f a;
    a.x = half ? (-2.0f * z1) : (-2.0f * x1);
    a.y = half ? 1.0f         : (-2.0f * y1);

    // C accumulator = n1 broadcast matching D layout: (v, lane) -> row v+8*(lane/16)
    v8f cvec;
#pragma unroll
    for (int v = 0; v < 8; ++v)
        cvec[v] = __shfl(n1_lane, v + 8 * half, 32);

    for (int m = tid; m < NPTS; m += 256) s_colmin[m] = 0x7F7FFFFFu;

    float rowmin[8];
#pragma unroll
    for (int v = 0; v < 8; ++v) rowmin[v] = FLT_MAX;

    for (int c = 0; c < NPTS / MCHUNK; ++c) {
        __syncthreads();
        // cooperative stage of p2 chunk (+norms) into LDS
        for (int i = tid; i < MCHUNK; i += 256) {
            const float* q = p2 + ((size_t)b * NPTS + (size_t)(c * MCHUNK + i)) * 3;
            float x = q[0], y = q[1], z = q[2];
            s_p2[i] = make_float4(x, y, z, x * x + y * y + z * z);
        }
        __syncthreads();

        // B column N = col; K-pair chosen by half. Stride per tile = 16 float4s.
        const float* spBase = ((const float*)&s_p2[col]) + 2 * half;

        // distance-PF software pipeline in a rotating register buffer.
        // Prefetch is UNGUARDED: overrun reads stay inside this workgroup's
        // 64KB LDS allocation and are never consumed.
        v2f buf[PF];
#pragma unroll
        for (int j = 0; j < PF; ++j)
            buf[j] = *(const v2f*)(spBase + (size_t)j * 64);

        for (int t = 0; t < NTILES; t += PF) {
#pragma unroll
            for (int j = 0; j < PF; ++j) {
                v2f bcur = buf[j];
                buf[j] = *(const v2f*)(spBase + (size_t)(t + PF + j) * 64);

                // full squared distances for this 16x16 tile
                v8f h = __builtin_amdgcn_wmma_f32_16x16x4_f32(
                            false, a, false, bcur, (short)0, cvec, false, false);

#pragma unroll
                for (int v = 0; v < 8; ++v)
                    rowmin[v] = fminf(rowmin[v], h[v]);

                float cmin = fminf(fminf(fminf(h[0], h[1]), fminf(h[2], h[3])),
                                   fminf(fminf(h[4], h[5]), fminf(h[6], h[7])));
                // all 32 lanes hit the column slot; ds_min merges the row-halves
                atomicMin(&s_colmin[c * MCHUNK + (t + j) * 16 + col],
                          __float_as_uint(fmaxf(cmin, 0.0f)));
            }
        }
    }

    // ---- finish row mins: reduce across the 16 lanes of each half ----
#pragma unroll
    for (int v = 0; v < 8; ++v) {
        float r = rowmin[v];
        r = fminf(r, __shfl_xor(r, 1, 32));
        r = fminf(r, __shfl_xor(r, 2, 32));
        r = fminf(r, __shfl_xor(r, 4, 32));
        r = fminf(r, __shfl_xor(r, 8, 32));
        rowmin[v] = r;
    }
    if (col == 0) {  // lanes 0 (rows 0..7) and 16 (rows 8..15)
#pragma unroll
        for (int v = 0; v < 8; ++v) {
            int row = nbase + v + 8 * half;
            g_rowmin[(size_t)b * NPTS + row] = fmaxf(rowmin[v], 0.0f);
        }
    }

    __syncthreads();
    for (int m = tid; m < NPTS; m += 256)
        atomicMin(&g_colmin[(size_t)b * NPTS + m], s_colmin[m]);
}

// ---------------- final: mean of all mins -> scalar ------------------------
__global__ __launch_bounds__(256) void chamfer_reduce(const float*    __restrict__ g_rowmin,
                                                      const unsigned* __restrict__ g_colmin,
                                                      float*          __restrict__ out) {
    __shared__ float red[256];
    float s = 0.0f;
    for (int i = threadIdx.x; i < BATCH * NPTS; i += 256)
        s += g_rowmin[i] + __uint_as_float(g_colmin[i]);
    red[threadIdx.x] = s;
    __syncthreads();
    for (int off = 128; off > 0; off >>= 1) {
        if (threadIdx.x < off) red[threadIdx.x] += red[threadIdx.x + off];
        __syncthreads();
    }
    if (threadIdx.x == 0)
        out[0] = red[0] / (float)(BATCH * NPTS);
}

extern "C" void kernel_launch(void* const* d_in, const int* in_sizes, int n_in,
                              void* d_out, int out_size, void* d_ws, size_t ws_size,
                              hipStream_t stream) {
    const float* p1 = (const float*)d_in[0];
    const float* p2 = (const float*)d_in[1];
    float* out = (float*)d_out;

    // workspace layout: [0, 128KB) col mins (uint bits), [128KB, 256KB) row mins (float)
    unsigned* g_colmin = (unsigned*)d_ws;
    float*    g_rowmin = (float*)((char*)d_ws + (size_t)BATCH * NPTS * sizeof(unsigned));

    chamfer_init<<<(BATCH * NPTS + 255) / 256, 256, 0, stream>>>(g_colmin);
    chamfer_main<<<BATCH * (NPTS / NROWS_PER_WG), 256, 0, stream>>>(p1, p2, g_rowmin, g_colmin);
    chamfer_reduce<<<1, 256, 0, stream>>>(g_rowmin, g_colmin, out);
}